// GCNModel_31602369364021
// MI455X (gfx1250) — compile-verified
//
#include <hip/hip_runtime.h>
#include <hip/hip_bf16.h>

typedef __attribute__((ext_vector_type(2))) float v2f;
typedef __attribute__((ext_vector_type(8))) float v8f;

#define D 64

__device__ __forceinline__ void atomAddF(float* p, float v) {
    // hardware global_atomic_add_f32 (no CAS loop)
    unsafeAtomicAdd(p, v);
}

__global__ void k_fill(float* __restrict__ p, int n, float v) {
    int t = blockIdx.x * blockDim.x + threadIdx.x;
    if (t < n) p[t] = v;
}

__global__ void k_deg(const int* __restrict__ dst, float* __restrict__ deg, int nedges) {
    int e = blockIdx.x * blockDim.x + threadIdx.x;
    if (e < nedges) atomAddF(&deg[dst[e]], 1.0f);
}

__global__ void k_rsqrt(const float* __restrict__ deg, float* __restrict__ dis, int n) {
    int t = blockIdx.x * blockDim.x + threadIdx.x;
    if (t < n) dis[t] = __frsqrt_rn(deg[t]);
}

// Y[nrows x 64] = X[nrows x 64] @ W[64 x 64], full fp32 via V_WMMA_F32_16X16X4_F32.
// One wave per 16x16 output tile; 16 chained K=4 WMMA steps.
__global__ __launch_bounds__(256) void k_gemm64(const float* __restrict__ X,
                                                const float* __restrict__ W,
                                                float* __restrict__ Y,
                                                int nrows) {
    const int lane      = threadIdx.x & 31;
    const int wave      = blockIdx.x * 8 + (threadIdx.x >> 5);
    const int total     = (nrows >> 4) * 4;          // row tiles * 4 col tiles
    if (wave >= total) return;                        // wave-uniform exit
    const int row_tile  = wave >> 2;
    const int col_tile  = wave & 3;
    const int m         = lane & 15;                  // M (for A) / N (for B,C)
    const int khalf     = lane >> 4;                  // lanes 16-31 hold K+2,K+3
    const int row       = (row_tile << 4) + m;
    const int col       = (col_tile << 4) + m;

    v8f c = {0.f, 0.f, 0.f, 0.f, 0.f, 0.f, 0.f, 0.f};
#pragma unroll
    for (int k = 0; k < D; k += 4) {
        const int ka = k + (khalf << 1);
        v2f a, b;
        a.x = X[row * D + ka];            // A 16x4: lane half selects K pair
        a.y = X[row * D + ka + 1];
        b.x = W[ka * D + col];            // B 4x16: row-striped across VGPRs
        b.y = W[(ka + 1) * D + col];
        c = __builtin_amdgcn_wmma_f32_16x16x4_f32(
                /*neg_a=*/false, a, /*neg_b=*/false, b,
                /*c_mod=*/(short)0, c, /*reuse_a=*/false, /*reuse_b=*/false);
    }
#pragma unroll
    for (int j = 0; j < 8; ++j) {
        const int r = (row_tile << 4) + j + (khalf << 3);  // VGPR j: M=j / M=j+8
        Y[r * D + col] = c[j];
    }
}

// One wave per edge; each lane handles 2 channels (float2 load + 2 atomic fadds).
__global__ __launch_bounds__(256) void k_scatter(const float* __restrict__ H,
                                                 const int* __restrict__ src,
                                                 const int* __restrict__ dst,
                                                 const float* __restrict__ dis,
                                                 float* __restrict__ agg,
                                                 int nedges) {
    const long long t = (long long)blockIdx.x * blockDim.x + threadIdx.x;
    const int e = (int)(t >> 5);
    if (e >= nedges) return;
    const int lane = (int)(t & 31);
    const int s = src[e];
    const int d = dst[e];
    const float norm = dis[s] * dis[d];
    const int c = lane << 1;
    const float2 h2 = *(const float2*)(H + s * D + c);
    float* o = agg + d * D + c;
    atomAddF(o, h2.x * norm);
    atomAddF(o + 1, h2.y * norm);
}

// agg = relu(agg + hlin * dis^2 (self-loop) + bias), in place.
__global__ void k_sl_bias_relu(float* __restrict__ agg, const float* __restrict__ hlin,
                               const float* __restrict__ dis, const float* __restrict__ bias,
                               int n) {
    int t = blockIdx.x * blockDim.x + threadIdx.x;
    if (t >= n * D) return;
    int i = t >> 6, c = t & 63;
    float ds = dis[i];
    float v = agg[t] + hlin[t] * ds * ds + bias[c];
    agg[t] = fmaxf(v, 0.0f);
}

__global__ void k_pool(const float* __restrict__ H, const int* __restrict__ batch,
                       float* __restrict__ psum, float* __restrict__ cnt, int n) {
    int t = blockIdx.x * blockDim.x + threadIdx.x;
    if (t >= n * D) return;
    int i = t >> 6, c = t & 63;
    int g = batch[i];
    atomAddF(&psum[g * D + c], H[t]);
    if (c == 0) atomAddF(&cnt[g], 1.0f);
}

__global__ void k_head(const float* __restrict__ psum, const float* __restrict__ cnt,
                       const float* __restrict__ Wfc, const float* __restrict__ bfc,
                       float* __restrict__ out, int ngraphs, int ncls) {
    int t = blockIdx.x * blockDim.x + threadIdx.x;
    if (t >= ngraphs * ncls) return;
    int g = t / ncls, cl = t % ncls;
    float inv = 1.0f / fmaxf(cnt[g], 1.0f);
    float acc = bfc[cl];
#pragma unroll
    for (int k = 0; k < D; ++k)
        acc = fmaf(psum[g * D + k] * inv, Wfc[k * ncls + cl], acc);
    out[t] = acc;
}

extern "C" void kernel_launch(void* const* d_in, const int* in_sizes, int n_in,
                              void* d_out, int out_size, void* d_ws, size_t ws_size,
                              hipStream_t stream) {
    const float* x   = (const float*)d_in[0];
    const int*   ei  = (const int*)d_in[1];
    const int*   bat = (const int*)d_in[2];
    const float* W1  = (const float*)d_in[3];
    const float* b1  = (const float*)d_in[4];
    const float* W2  = (const float*)d_in[5];
    const float* b2  = (const float*)d_in[6];
    const float* Wfc = (const float*)d_in[7];
    const float* bfc = (const float*)d_in[8];
    float* out = (float*)d_out;

    const int n    = in_sizes[0] / D;   // 50000 nodes
    const int ne   = in_sizes[1] / 2;   // 1.6M edges
    const int ncls = in_sizes[8];       // 10
    const int ng   = out_size / ncls;   // 64 graphs

    const int* src = ei;
    const int* dst = ei + ne;

    char* ws = (char*)d_ws;
    const size_t feat = (size_t)n * D * sizeof(float);
    float* bufA = (float*)ws; ws += feat;                       // lin output
    float* bufB = (float*)ws; ws += feat;                       // agg1 / H1
    float* bufC = (float*)ws; ws += feat;                       // agg2 / H2
    float* deg  = (float*)ws; ws += (size_t)n * sizeof(float);
    float* dis  = (float*)ws; ws += (size_t)n * sizeof(float);
    float* psum = (float*)ws; ws += (size_t)ng * D * sizeof(float);
    float* cnt  = (float*)ws; ws += (size_t)ng * sizeof(float); // contiguous after psum

    const int B = 256;
    const int nodeBlocks = (n + B - 1) / B;
    const int featBlocks = (n * D + B - 1) / B;
    const int edgeBlocks = (ne + B - 1) / B;
    const int gemmBlocks = ((n / 16) * 4 + 7) / 8;              // 8 waves / block
    const long long st   = (long long)ne * 32;
    const int scatBlocks = (int)((st + B - 1) / B);

    // normalization: deg (with self-loops) -> deg^{-1/2}
    k_fill<<<nodeBlocks, B, 0, stream>>>(deg, n, 1.0f);
    k_deg<<<edgeBlocks, B, 0, stream>>>(dst, deg, ne);
    k_rsqrt<<<nodeBlocks, B, 0, stream>>>(deg, dis, n);

    // layer 1
    k_gemm64<<<gemmBlocks, B, 0, stream>>>(x, W1, bufA, n);
    k_fill<<<featBlocks, B, 0, stream>>>(bufB, n * D, 0.0f);
    k_scatter<<<scatBlocks, B, 0, stream>>>(bufA, src, dst, dis, bufB, ne);
    k_sl_bias_relu<<<featBlocks, B, 0, stream>>>(bufB, bufA, dis, b1, n);

    // layer 2
    k_gemm64<<<gemmBlocks, B, 0, stream>>>(bufB, W2, bufA, n);
    k_fill<<<featBlocks, B, 0, stream>>>(bufC, n * D, 0.0f);
    k_scatter<<<scatBlocks, B, 0, stream>>>(bufA, src, dst, dis, bufC, ne);
    k_sl_bias_relu<<<featBlocks, B, 0, stream>>>(bufC, bufA, dis, b2, n);

    // global mean pool + classifier head
    k_fill<<<(ng * D + ng + B - 1) / B, B, 0, stream>>>(psum, ng * D + ng, 0.0f);
    k_pool<<<featBlocks, B, 0, stream>>>(bufC, bat, psum, cnt, n);
    k_head<<<(ng * ncls + B - 1) / B, B, 0, stream>>>(psum, cnt, Wfc, bfc, out, ng, ncls);
}